// GCNNet_73452530696689
// MI455X (gfx1250) — compile-verified
//
#include <hip/hip_runtime.h>
#include <hip/hip_bf16.h>
#include <math.h>

typedef __attribute__((ext_vector_type(2))) float v2f;
typedef __attribute__((ext_vector_type(8))) float v8f;

#define EMB 256
#define HID 512
#define LDSA_STRIDE 260   // pad 256 -> 260 floats so 16 rows hit distinct LDS banks

// ---------------------------------------------------------------------------
// GEMM1: hw1[n, 512] = embed_table[x[n], 256] @ W1[256, 512]  (fp32 WMMA)
// Block = 256 threads (8 waves). Block tile = 16 rows x 128 cols.
// Wave w owns 16x16 tile at col0 = blockIdx.y*128 + w*16.
// K loop: 64 steps of V_WMMA_F32_16X16X4_F32.
// ---------------------------------------------------------------------------
__global__ __launch_bounds__(256) void gcn_gemm1_wmma(
    const int* __restrict__ x,
    const float* __restrict__ emb,   // [N, 256]
    const float* __restrict__ W1,    // [256, 512]
    float* __restrict__ hw1,         // [N, 512]
    int n_nodes)
{
    __shared__ float lds_a[16][LDSA_STRIDE];   // 16 rows x 256 K (padded)

    const int tid  = threadIdx.x;
    const int wave = tid >> 5;
    const int lane = tid & 31;
    const int row0 = blockIdx.x * 16;
    const int col0 = blockIdx.y * 128 + wave * 16;

    // Cooperatively stage the 16x256 A tile into LDS (float4 granularity).
    for (int i = tid; i < 16 * (EMB / 4); i += 256) {
        const int r  = i >> 6;            // / (EMB/4) = /64
        const int c4 = i & 63;            // float4 index within row
        float4 v = make_float4(0.f, 0.f, 0.f, 0.f);
        const int row = row0 + r;
        if (row < n_nodes) {
            const int srow = x[row];      // embedding lookup (identity here)
            v = ((const float4*)(emb + (size_t)srow * EMB))[c4];
        }
        ((float4*)&lds_a[r][0])[c4] = v;
    }
    __syncthreads();

    const int half = lane >> 4;   // 0: K offset {0,1}; 1: K offset {2,3}
    const int l15  = lane & 15;

    v8f acc = {};
    #pragma unroll 4
    for (int k0 = 0; k0 < EMB; k0 += 4) {
        // A fragment (16x4 f32): lane l15 = row M, VGPR0/1 = K pair
        v2f a;
        const float* ap = &lds_a[l15][k0 + 2 * half];
        a.x = ap[0];
        a.y = ap[1];
        // B fragment (4x16 f32): lane l15 = col N, half selects K pair
        const int kb = k0 + 2 * half;
        const float* bp = W1 + (size_t)kb * HID + col0 + l15;
        v2f b;
        b.x = bp[0];
        b.y = bp[HID];
        acc = __builtin_amdgcn_wmma_f32_16x16x4_f32(
            /*neg_a=*/false, a, /*neg_b=*/false, b,
            /*c_mod=*/(short)0, acc, /*reuse_a=*/false, /*reuse_b=*/false);
    }

    // D tile: VGPR v holds (M = v + 8*half, N = l15)
    #pragma unroll
    for (int v = 0; v < 8; ++v) {
        const int row = row0 + v + 8 * half;
        if (row < n_nodes)
            hw1[(size_t)row * HID + col0 + l15] = acc[v];
    }
}

// ---------------------------------------------------------------------------
// Init agg1[n, 512] = b1[f]  (bias pre-loaded so scatter lands on top of it)
// ---------------------------------------------------------------------------
__global__ __launch_bounds__(256) void init_agg1(
    float* __restrict__ agg1, const float* __restrict__ b1, int n_nodes)
{
    const long long t = (long long)blockIdx.x * blockDim.x + threadIdx.x;
    const long long total = (long long)n_nodes * (HID / 4);
    if (t >= total) return;
    const int c4 = (int)(t & 127);           // 512/4 = 128 chunks per row
    const float4 bv = ((const float4*)b1)[c4];
    ((float4*)agg1)[t] = bv;
}

// ---------------------------------------------------------------------------
// Scatter layer 1: agg1[dst[e], f] += ew[e] * hw1[src[e], f]
// Block = 256 threads = 2 edges x 128 float4-chunks. Edge scalars are
// wave-uniform (scalarized); gathers/atomics are coalesced float4 spans.
// ---------------------------------------------------------------------------
__global__ __launch_bounds__(256) void scatter1(
    const int* __restrict__ src, const int* __restrict__ dst,
    const float* __restrict__ ew,
    const float* __restrict__ hw1, float* __restrict__ agg1, int E)
{
    const int e = blockIdx.x * 2 + (threadIdx.x >> 7);
    if (e >= E) return;
    const int f = (threadIdx.x & 127) << 2;
    const int s = src[e];
    const int d = dst[e];
    const float w = ew[e];
    const float4 v = *(const float4*)(hw1 + (size_t)s * HID + f);
    float* p = agg1 + (size_t)d * HID + f;
    unsafeAtomicAdd(p + 0, w * v.x);
    unsafeAtomicAdd(p + 1, w * v.y);
    unsafeAtomicAdd(p + 2, w * v.z);
    unsafeAtomicAdd(p + 3, w * v.w);
}

// ---------------------------------------------------------------------------
// GEMM2 (+ ReLU on input): hw2[n, 2] = relu(agg1[n, :]) @ W2[512, 2]
// One wave per node: strided 512-dot, shuffle reduction. N=2 => WMMA wasteful.
// ---------------------------------------------------------------------------
__global__ __launch_bounds__(256) void gemm2_relu(
    const float* __restrict__ agg1, const float* __restrict__ W2,
    float* __restrict__ hw2, int n_nodes)
{
    const int gwave = (blockIdx.x * 256 + threadIdx.x) >> 5;
    const int lane  = threadIdx.x & 31;
    if (gwave >= n_nodes) return;
    const float* row = agg1 + (size_t)gwave * HID;
    float a0 = 0.f, a1 = 0.f;
    #pragma unroll 4
    for (int k = lane; k < HID; k += 32) {
        const float h = fmaxf(row[k], 0.f);
        a0 += h * W2[2 * k + 0];
        a1 += h * W2[2 * k + 1];
    }
    #pragma unroll
    for (int off = 16; off > 0; off >>= 1) {
        a0 += __shfl_down(a0, off, 32);
        a1 += __shfl_down(a1, off, 32);
    }
    if (lane == 0) {
        hw2[2 * gwave + 0] = a0;
        hw2[2 * gwave + 1] = a1;
    }
}

// ---------------------------------------------------------------------------
// Init agg2[n, 2] = b2
// ---------------------------------------------------------------------------
__global__ __launch_bounds__(256) void init_agg2(
    float* __restrict__ agg2, const float* __restrict__ b2, int n_nodes)
{
    const int i = blockIdx.x * blockDim.x + threadIdx.x;
    if (i >= n_nodes) return;
    agg2[2 * i + 0] = b2[0];
    agg2[2 * i + 1] = b2[1];
}

// ---------------------------------------------------------------------------
// Scatter layer 2: agg2[dst[e], c] += ew[e] * hw2[src[e], c], c in {0,1}
// ---------------------------------------------------------------------------
__global__ __launch_bounds__(256) void scatter2(
    const int* __restrict__ src, const int* __restrict__ dst,
    const float* __restrict__ ew,
    const float* __restrict__ hw2, float* __restrict__ agg2, int E)
{
    const int e = blockIdx.x * blockDim.x + threadIdx.x;
    if (e >= E) return;
    const int s = src[e];
    const int d = dst[e];
    const float w = ew[e];
    const float2 v = *(const float2*)(hw2 + 2 * (size_t)s);
    unsafeAtomicAdd(agg2 + 2 * (size_t)d + 0, w * v.x);
    unsafeAtomicAdd(agg2 + 2 * (size_t)d + 1, w * v.y);
}

// ---------------------------------------------------------------------------
// Final: out[n, :] = log_softmax(relu(agg2[n, :]))   (bias already in agg2)
// ---------------------------------------------------------------------------
__global__ __launch_bounds__(256) void final_logsoftmax(
    const float* __restrict__ agg2, float* __restrict__ out, int n_nodes)
{
    const int i = blockIdx.x * blockDim.x + threadIdx.x;
    if (i >= n_nodes) return;
    const float z0 = fmaxf(agg2[2 * i + 0], 0.f);
    const float z1 = fmaxf(agg2[2 * i + 1], 0.f);
    const float m  = fmaxf(z0, z1);
    const float lse = m + logf(expf(z0 - m) + expf(z1 - m));
    out[2 * i + 0] = z0 - lse;
    out[2 * i + 1] = z1 - lse;
}

// ---------------------------------------------------------------------------
extern "C" void kernel_launch(void* const* d_in, const int* in_sizes, int n_in,
                              void* d_out, int out_size, void* d_ws, size_t ws_size,
                              hipStream_t stream) {
    const int*   x     = (const int*)  d_in[0];           // [N]
    const int*   eidx  = (const int*)  d_in[1];           // [2, E]
    const float* eattr = (const float*)d_in[2];           // [E]
    const float* emb   = (const float*)d_in[3];           // [N, 256]
    const float* W1    = (const float*)d_in[4];           // [256, 512]
    const float* b1    = (const float*)d_in[5];           // [512]
    const float* W2    = (const float*)d_in[6];           // [512, 2]
    const float* b2    = (const float*)d_in[7];           // [2]
    float*       out   = (float*)d_out;                   // [N, 2]

    const int N = in_sizes[0];
    const int E = in_sizes[2];
    const int* src = eidx;
    const int* dst = eidx + E;

    // Workspace carve-up
    char* ws = (char*)d_ws;
    float* hw1  = (float*)ws;                                  // N*512 f32
    float* agg1 = (float*)(ws + (size_t)N * HID * sizeof(float));
    float* hw2  = (float*)(ws + 2 * (size_t)N * HID * sizeof(float));
    float* agg2 = (float*)(ws + 2 * (size_t)N * HID * sizeof(float)
                              + (size_t)N * 2 * sizeof(float));

    // 1) GEMM1 via fp32 WMMA: hw1 = emb[x] @ W1
    {
        dim3 grid((N + 15) / 16, HID / 128);
        gcn_gemm1_wmma<<<grid, 256, 0, stream>>>(x, emb, W1, hw1, N);
    }
    // 2) agg1 = b1 (broadcast)
    {
        const long long total = (long long)N * (HID / 4);
        init_agg1<<<(unsigned)((total + 255) / 256), 256, 0, stream>>>(agg1, b1, N);
    }
    // 3) scatter-add layer 1
    scatter1<<<(E + 1) / 2, 256, 0, stream>>>(src, dst, eattr, hw1, agg1, E);
    // 4) hw2 = relu(agg1) @ W2  (wave-per-node)
    gemm2_relu<<<(N * 32 + 255) / 256, 256, 0, stream>>>(agg1, W2, hw2, N);
    // 5) agg2 = b2
    init_agg2<<<(N + 255) / 256, 256, 0, stream>>>(agg2, b2, N);
    // 6) scatter-add layer 2
    scatter2<<<(E + 255) / 256, 256, 0, stream>>>(src, dst, eattr, hw2, agg2, E);
    // 7) out = log_softmax(relu(agg2))
    final_logsoftmax<<<(N + 255) / 256, 256, 0, stream>>>(agg2, out, N);
}